// PointSpatioTemporalCorrelation_71751723647261
// MI455X (gfx1250) — compile-verified
//
#include <hip/hip_runtime.h>
#include <math.h>

#define NPTS   2048
#define CIN    64
#define COUT   128
#define FANIN  195          // Cin + Cout + 3
#define BATCH  8
#define NSAMP  32
#define RAD2   (0.15f * 0.15f)
#define GROW   256          // workspace row: [0..127]=TU, [128..255]=V

typedef float v2f __attribute__((ext_vector_type(2)));
typedef float v8f __attribute__((ext_vector_type(8)));
typedef int   v4i __attribute__((vector_size(16)));   // matches builtin param type

__device__ __forceinline__ v8f wmma4(v2f a, v2f b, v8f c) {
  // V_WMMA_F32_16X16X4_F32: D(16x16,f32) = A(16x4) * B(4x16) + C, exact fp32
  return __builtin_amdgcn_wmma_f32_16x16x4_f32(false, a, false, b, (short)0, c,
                                               false, false);
}

// ---------------------------------------------------------------------------
// Kernel 1: per-point GEMMs via WMMA f32 16x16x4.
//   G[b][n][o]       = TU = W[:,0:128]@S2[b,:,n] + W[:,192:195]@P2[b,n,:]   (o<128)
//   G[b][n][128+o]   = V  = W[:,128:192]@X1[b,:,n] - W[:,192:195]@P1[b,n,:] + bias[o]
// One wave computes one 16(points) x 16(outputs) tile.
// A layout (doc, 32-bit 16x4): m = lane&15, VGPR0/1 hold K = kh,kh+1 with
// kh = (lane>>4)*2.  B 4x16 assumed mirrored row-striping: same kh per lane half.
// D layout (doc): VGPR i -> point = i + 8*(lane>=16), out = lane&15.
// ---------------------------------------------------------------------------
__global__ __launch_bounds__(128, 4)
void pstc_gemm(const float* __restrict__ P1, const float* __restrict__ P2,
               const float* __restrict__ X1, const float* __restrict__ S2,
               const float* __restrict__ W,  const float* __restrict__ bias,
               float* __restrict__ G)
{
  const int lane = threadIdx.x & 31;
  const int wid  = blockIdx.x * (blockDim.x >> 5) + (threadIdx.x >> 5);
  const int tilesPerBatch = (NPTS / 16) * 16;            // 128 ntiles * 16 otiles
  const int b     = wid / tilesPerBatch;
  const int rem   = wid % tilesPerBatch;
  const int n0    = (rem >> 4) * 16;                     // point tile base
  const int o0g   = (rem & 15) * 16;                     // 0..255 output tile base
  const int m     = lane & 15;                           // A: point row / B,D: column
  const int kh    = (lane >> 4) << 1;                    // 0 or 2
  const int og    = o0g + m;                             // global G column 0..255

  v8f acc = {};

  if (o0g < COUT) {
    // ---- TU half: S2 part (K=0..127) + P2 coords (K=192..194) ----
    const int o = og;                                    // output channel
    const float* S2b = S2 + (size_t)b * COUT * NPTS;
    const float* Wo  = W + (size_t)o * FANIN;
#pragma unroll 4
    for (int k0 = 0; k0 < COUT; k0 += 4) {
      v2f a, bm;
      a.x  = S2b[(size_t)(k0 + kh)     * NPTS + n0 + m];
      a.y  = S2b[(size_t)(k0 + kh + 1) * NPTS + n0 + m];
      bm.x = Wo[k0 + kh];
      bm.y = Wo[k0 + kh + 1];
      acc = wmma4(a, bm, acc);
    }
    {
      const float* p2 = P2 + (size_t)(b * NPTS + n0 + m) * 3;
      v2f a, bm;
      a.x  = p2[kh];                                     // kh in {0,2} -> valid
      a.y  = (kh + 1 < 3) ? p2[kh + 1] : 0.0f;           // K=3 padded with 0
      bm.x = Wo[192 + kh];
      bm.y = (kh + 1 < 3) ? Wo[192 + kh + 1] : 0.0f;
      acc = wmma4(a, bm, acc);
    }
  } else {
    // ---- V half: X1 part (K=128..191) - W_disp@P1 + bias ----
    const int o = og - COUT;
    const float* X1b = X1 + (size_t)b * CIN * NPTS;
    const float* Wo  = W + (size_t)o * FANIN;
    const float bv   = bias[o];
#pragma unroll
    for (int i = 0; i < 8; ++i) acc[i] = bv;             // bias in C (all point rows)
#pragma unroll 4
    for (int k0 = 0; k0 < CIN; k0 += 4) {
      v2f a, bm;
      a.x  = X1b[(size_t)(k0 + kh)     * NPTS + n0 + m];
      a.y  = X1b[(size_t)(k0 + kh + 1) * NPTS + n0 + m];
      bm.x = Wo[COUT + k0 + kh];
      bm.y = Wo[COUT + k0 + kh + 1];
      acc = wmma4(a, bm, acc);
    }
    {
      const float* p1 = P1 + (size_t)(b * NPTS + n0 + m) * 3;
      v2f a, bm;
      a.x  = p1[kh];
      a.y  = (kh + 1 < 3) ? p1[kh + 1] : 0.0f;
      bm.x = -Wo[192 + kh];                              // minus: disp = P2 - P1
      bm.y = (kh + 1 < 3) ? -Wo[192 + kh + 1] : 0.0f;
      acc = wmma4(a, bm, acc);
    }
  }

  // Store D: VGPR i -> point prow+i, column og
  float* Gp = G + (size_t)(b * NPTS + n0) * GROW + og;
  const int prow = (lane >> 4) << 3;                     // 0 or 8
#pragma unroll
  for (int i = 0; i < 8; ++i)
    Gp[(size_t)(prow + i) * GROW] = acc[i];
}

// ---------------------------------------------------------------------------
// Kernel 2: fused ball-query + gather + max + relu.
// One wave per query; support coords staged in LDS via CDNA5 async
// global->LDS DMA (ASYNCcnt path, no VGPR round-trip); first-32 ascending
// hits via wave32 ballot + ctz (reference's sort-then-truncate == first 32
// in index order).  Each lane owns 4 output channels, so each TU gather is
// one fully coalesced 512B float4 row out of the L2-resident table.
// ---------------------------------------------------------------------------
#define QPB 32   // queries per block (8 waves x 4)

#if __has_builtin(__builtin_amdgcn_global_load_async_to_lds_b128)
#define HAVE_ASYNC_LDS 1
#endif

__global__ __launch_bounds__(256, 2)
void pstc_gather(const float* __restrict__ P1, const float* __restrict__ P2,
                 const float* __restrict__ G, float* __restrict__ out)
{
  __shared__ float sp[NPTS * 3];                         // interleaved x,y,z (24 KB)

  const int b     = blockIdx.x / (NPTS / QPB);
  const int qbase = (blockIdx.x % (NPTS / QPB)) * QPB;

  const float* P2b = P2 + (size_t)b * NPTS * 3;
#ifdef HAVE_ASYNC_LDS
  // 24 KB as 1536 x 16B async DMA transfers, 6 per thread, no VGPR staging.
  for (int i = threadIdx.x; i < (NPTS * 3) / 4; i += blockDim.x) {
    __builtin_amdgcn_global_load_async_to_lds_b128(
        (v4i*)(P2b + i * 4), (v4i*)(sp + i * 4), 0, 0);
  }
#if __has_builtin(__builtin_amdgcn_s_wait_asynccnt)
  __builtin_amdgcn_s_wait_asynccnt(0);
#else
  asm volatile("s_wait_asynccnt 0x0" ::: "memory");
#endif
#else
  for (int i = threadIdx.x; i < NPTS * 3; i += blockDim.x)
    sp[i] = P2b[i];
#endif
  __syncthreads();

  const int lane = threadIdx.x & 31;
  const int wv   = threadIdx.x >> 5;
  const float4* G4 = (const float4*)G;                   // 64 float4 per point row

  for (int q = wv; q < QPB; q += 8) {
    const int n = qbase + q;
    const float* p1 = P1 + (size_t)(b * NPTS + n) * 3;
    const float qx = p1[0], qy = p1[1], qz = p1[2];

    float4 mx;
    mx.x = mx.y = mx.z = mx.w = -3.402823466e38f;
    int hits = 0;

    for (int s0 = 0; s0 < NPTS && hits < NSAMP; s0 += 32) {
      const int s = s0 + lane;
      const float dx = qx - sp[3 * s + 0];
      const float dy = qy - sp[3 * s + 1];
      const float dz = qz - sp[3 * s + 2];
      const float d2 = dx * dx + dy * dy + dz * dz;
      unsigned mask = __builtin_amdgcn_ballot_w32(d2 < RAD2);
      while (mask && hits < NSAMP) {
        const int j = __builtin_ctz(mask);               // ascending index order
        mask &= mask - 1;
        const float4 t = G4[(size_t)(b * NPTS + s0 + j) * 64 + lane];
        mx.x = fmaxf(mx.x, t.x);
        mx.y = fmaxf(mx.y, t.y);
        mx.z = fmaxf(mx.z, t.z);
        mx.w = fmaxf(mx.w, t.w);
        ++hits;
      }
    }
    if (hits == 0)                                       // reference: idx -> 0
      mx = G4[(size_t)(b * NPTS) * 64 + lane];

    const float4 v = G4[(size_t)(b * NPTS + n) * 64 + 32 + lane];  // V half
    float4 r;
    r.x = fmaxf(mx.x + v.x, 0.0f);
    r.y = fmaxf(mx.y + v.y, 0.0f);
    r.z = fmaxf(mx.z + v.z, 0.0f);
    r.w = fmaxf(mx.w + v.w, 0.0f);

    float* ob = out + (size_t)b * COUT * NPTS + n;       // out[b][o][n]
    const int o = lane * 4;
    ob[(size_t)(o + 0) * NPTS] = r.x;
    ob[(size_t)(o + 1) * NPTS] = r.y;
    ob[(size_t)(o + 2) * NPTS] = r.z;
    ob[(size_t)(o + 3) * NPTS] = r.w;
  }
}

// ---------------------------------------------------------------------------
extern "C" void kernel_launch(void* const* d_in, const int* in_sizes, int n_in,
                              void* d_out, int out_size, void* d_ws, size_t ws_size,
                              hipStream_t stream)
{
  const float* P1   = (const float*)d_in[0];   // (B, N, 3)
  const float* P2   = (const float*)d_in[1];   // (B, N, 3)
  const float* X1   = (const float*)d_in[2];   // (B, 64, N)
  const float* S2   = (const float*)d_in[3];   // (B, 128, N)
  const float* W    = (const float*)d_in[4];   // (128, 195)
  const float* bias = (const float*)d_in[5];   // (128,)
  float* out = (float*)d_out;                  // (B, 128, N)
  float* G   = (float*)d_ws;                   // (B, N, 256) = 16 MB scratch

  // 16384 tile-waves, 4 waves (128 threads) per block
  const int waves = BATCH * (NPTS / 16) * 16;
  pstc_gemm<<<waves / 4, 128, 0, stream>>>(P1, P2, X1, S2, W, bias, G);

  pstc_gather<<<BATCH * (NPTS / QPB), 256, 0, stream>>>(P1, P2, G, out);
}